// GCN_63256278335621
// MI455X (gfx1250) — compile-verified
//
#include <hip/hip_runtime.h>

typedef float v2f __attribute__((ext_vector_type(2)));
typedef float v8f __attribute__((ext_vector_type(8)));

#ifndef GCN_F_IN
#define GCN_F_IN 128
#define GCN_HID  256
#endif

// ---------------- degree / normalization ----------------
__global__ __launch_bounds__(256) void deg_init(float* __restrict__ deg, int n) {
  int i = blockIdx.x * blockDim.x + threadIdx.x;
  if (i < n) deg[i] = 1.0f;  // self-loop contributes 1
}

__global__ __launch_bounds__(256) void deg_count(float* __restrict__ deg,
                                                 const long long* __restrict__ dst, int e) {
  int i = blockIdx.x * blockDim.x + threadIdx.x;
  if (i < e) atomicAdd(&deg[(int)dst[i]], 1.0f);
}

__global__ __launch_bounds__(256) void deg_rsqrt(float* __restrict__ deg, int n) {
  int i = blockIdx.x * blockDim.x + threadIdx.x;
  if (i < n) deg[i] = rsqrtf(deg[i]);  // deg >= 1 always
}

__global__ __launch_bounds__(256) void buf_zero4(float4* __restrict__ p, int n4) {
  int i = blockIdx.x * blockDim.x + threadIdx.x;
  if (i < n4) p[i] = make_float4(0.f, 0.f, 0.f, 0.f);
}

// ---------------- fp32 WMMA GEMM: H[r,:] = dinv[r] * (X @ W)[r,:] ----------------
// 4 waves/block; each wave owns a 16-row stripe and computes NTILES (=4)
// adjacent 16x16 output tiles, so each A fragment load feeds 4 WMMAs.
// W panel staged in LDS with K-pairs interleaved so every B fragment is one
// aligned ds_load_b64 into an even VGPR pair (no repack moves before WMMA):
//   Bs[((k>>1)*NB + c)*2 + (k&1)] = W[k][n0+c]
template <int KDIM, int NTILES>
__global__ __launch_bounds__(128) void gcn_gemm_wmma(
    const float* __restrict__ X,      // [M, KDIM]
    const float* __restrict__ W,      // [KDIM, Nf]
    const float* __restrict__ dinv,   // [M]
    float* __restrict__ H,            // [M, Nf]  (dinv-scaled product)
    int M, int Nf) {
  constexpr int NB = 16 * NTILES;
  __shared__ float Bs[KDIM * NB];

  const int tid  = threadIdx.x;
  const int wave = tid >> 5;
  const int lane = tid & 31;

  const int nblocksN = Nf / NB;
  const int bm = blockIdx.x / nblocksN;          // 64-row block
  const int n0 = (blockIdx.x % nblocksN) * NB;   // 64-col panel

  // Stage W[:, n0:n0+NB] into LDS in K-pair-interleaved layout
  for (int idx = tid; idx < KDIM * NB; idx += 128) {
    int k = idx / NB, c = idx % NB;
    Bs[(((k >> 1) * NB + c) << 1) | (k & 1)] = W[(size_t)k * Nf + n0 + c];
  }
  __syncthreads();

  const int m0 = bm * 64 + wave * 16;
  if (m0 >= M) return;                           // M % 16 == 0 -> full tiles

  const int col  = lane & 15;                    // N index within tile (and A row)
  const int koff = (lane >> 4) << 1;             // lanes 16-31 hold K+2,K+3

  const float* Arow = X + (size_t)(m0 + col) * KDIM;

  v8f acc[NTILES];
  #pragma unroll
  for (int t = 0; t < NTILES; ++t) acc[t] = (v8f){};

  for (int kb = 0; kb < KDIM; kb += 4) {
    // A 16x4: VGPR0 = K=(0|2), VGPR1 = K=(1|3); M = lane&15
    v2f a = *(const v2f*)(Arow + kb + koff);
    const int prow = ((kb + koff) >> 1) * NB;    // K-pair row in Bs
    #pragma unroll
    for (int t = 0; t < NTILES; ++t) {
      // B 4x16 fragment: lanes0-15 K={kb,kb+1}, lanes16-31 K={kb+2,kb+3}
      v2f b = *(const v2f*)&Bs[(prow + t * 16 + col) << 1];
      acc[t] = __builtin_amdgcn_wmma_f32_16x16x4_f32(
          /*neg_a=*/false, a, /*neg_b=*/false, b,
          /*c_mod=*/(short)0, acc[t], /*reuse_a=*/false, /*reuse_b=*/false);
    }
  }

  // C/D layout: VGPR v -> row m0 + v + (lane>=16 ? 8 : 0), col = lane&15
  const int rbase = m0 + ((lane >> 4) << 3);
  #pragma unroll
  for (int v = 0; v < 8; ++v) {
    const int r = rbase + v;
    const float dv = dinv[r];
    float* hrow = H + (size_t)r * Nf + n0 + col;
    #pragma unroll
    for (int t = 0; t < NTILES; ++t) hrow[t * 16] = acc[t][v] * dv;
  }
}

// ---------------- edge scatter: agg[dst,:] += H[src,:]  (one wave per edge) ----------------
__global__ __launch_bounds__(256) void gcn_scatter(
    const float* __restrict__ H, float* __restrict__ agg,
    const long long* __restrict__ src, const long long* __restrict__ dst,
    int E, int Nf) {
  // wave-uniform edge id -> scalar index loads + saddr-form gathers
  const int e = __builtin_amdgcn_readfirstlane(blockIdx.x * 8 + (threadIdx.x >> 5));
  const int lane = threadIdx.x & 31;
  if (e >= E) return;
  const int s = (int)src[e];
  const int d = (int)dst[e];
  const float4* __restrict__ hs = (const float4*)(H + (size_t)s * Nf);
  float* __restrict__ ag = agg + (size_t)d * Nf;
  const int nf4 = Nf >> 2;
  #pragma unroll 2
  for (int q = lane; q < nf4; q += 32) {
    float4 v = hs[q];
    int f = q << 2;
    atomicAdd(&ag[f + 0], v.x);
    atomicAdd(&ag[f + 1], v.y);
    atomicAdd(&ag[f + 2], v.z);
    atomicAdd(&ag[f + 3], v.w);
  }
}

// ---------------- finalize: out = act(dinv[r]*(agg + H_selfloop) + b) ----------------
__global__ __launch_bounds__(256) void gcn_finalize(
    const float4* __restrict__ agg, const float4* __restrict__ H,
    const float* __restrict__ dinv, const float4* __restrict__ bias,
    float4* __restrict__ out, int M, int Nf, int act /*0=relu, 1=relu+sigmoid*/) {
  const int nf4 = Nf >> 2;
  int i = blockIdx.x * blockDim.x + threadIdx.x;   // float4 index
  if (i >= M * nf4) return;
  int r = i / nf4;
  int f4 = i - r * nf4;
  const float dv = dinv[r];
  float4 a = agg[i], h = H[i], b = bias[f4];
  float4 o;
  o.x = fmaxf(dv * (a.x + h.x) + b.x, 0.f);
  o.y = fmaxf(dv * (a.y + h.y) + b.y, 0.f);
  o.z = fmaxf(dv * (a.z + h.z) + b.z, 0.f);
  o.w = fmaxf(dv * (a.w + h.w) + b.w, 0.f);
  if (act) {
    o.x = 1.f / (1.f + expf(-o.x));
    o.y = 1.f / (1.f + expf(-o.y));
    o.z = 1.f / (1.f + expf(-o.z));
    o.w = 1.f / (1.f + expf(-o.w));
  }
  out[i] = o;
}

// ---------------- host orchestration ----------------
extern "C" void kernel_launch(void* const* d_in, const int* in_sizes, int n_in,
                              void* d_out, int out_size, void* d_ws, size_t ws_size,
                              hipStream_t stream) {
  (void)n_in; (void)out_size; (void)ws_size;

  const float*     x    = (const float*)d_in[0];
  const long long* eidx = (const long long*)d_in[1];
  const float*     W1   = (const float*)d_in[2];
  const float*     b1   = (const float*)d_in[3];
  const float*     W2   = (const float*)d_in[4];
  const float*     b2   = (const float*)d_in[5];
  const float*     W3   = (const float*)d_in[6];
  const float*     b3   = (const float*)d_in[7];

  const int N = in_sizes[0] / GCN_F_IN;
  const int E = in_sizes[1] / 2;
  const long long* src = eidx;       // edge_index[0]
  const long long* dst = eidx + E;   // edge_index[1]

  // workspace layout: dinv | H (N*256) | AGG (N*256)
  float* ws   = (float*)d_ws;
  size_t off  = ((size_t)N + 63) & ~(size_t)63;
  float* dinv = ws;
  float* H    = ws + off;
  float* AGG  = H + (size_t)N * GCN_HID;
  float* OUT  = (float*)d_out;

  const int T = 256;
  // 1) symmetric normalization: dinv = rsqrt(1 + in_degree)
  deg_init <<<(N + T - 1) / T, T, 0, stream>>>(dinv, N);
  deg_count<<<(E + T - 1) / T, T, 0, stream>>>(dinv, dst, E);
  deg_rsqrt<<<(N + T - 1) / T, T, 0, stream>>>(dinv, N);

  const int mblk  = (N + 63) / 64;
  const int eblk  = (E + 7) / 8;                 // 8 waves (edges) per block

  // ---- layer 1: x[N,128] @ W1[128,256] ----
  gcn_gemm_wmma<GCN_F_IN, 4><<<mblk * (GCN_HID / 64), 128, 0, stream>>>(x, W1, dinv, H, N, GCN_HID);
  {
    int n4 = N * (GCN_HID / 4);
    buf_zero4<<<(n4 + T - 1) / T, T, 0, stream>>>((float4*)AGG, n4);
    gcn_scatter<<<eblk, T, 0, stream>>>(H, AGG, src, dst, E, GCN_HID);
    gcn_finalize<<<(n4 + T - 1) / T, T, 0, stream>>>((const float4*)AGG, (const float4*)H,
                                                     dinv, (const float4*)b1, (float4*)AGG,
                                                     N, GCN_HID, 0);
  }

  // ---- layer 2: act1[N,256] @ W2[256,256] ----
  gcn_gemm_wmma<GCN_HID, 4><<<mblk * (GCN_HID / 64), 128, 0, stream>>>(AGG, W2, dinv, H, N, GCN_HID);
  {
    int n4 = N * (GCN_HID / 4);
    buf_zero4<<<(n4 + T - 1) / T, T, 0, stream>>>((float4*)AGG, n4);
    gcn_scatter<<<eblk, T, 0, stream>>>(H, AGG, src, dst, E, GCN_HID);
    gcn_finalize<<<(n4 + T - 1) / T, T, 0, stream>>>((const float4*)AGG, (const float4*)H,
                                                     dinv, (const float4*)b2, (float4*)AGG,
                                                     N, GCN_HID, 0);
  }

  // ---- layer 3: act2[N,256] @ W3[256,128], sigmoid(relu(.)) -> d_out ----
  gcn_gemm_wmma<GCN_HID, 4><<<mblk * (GCN_F_IN / 64), 128, 0, stream>>>(AGG, W3, dinv, H, N, GCN_F_IN);
  {
    int n4 = N * (GCN_F_IN / 4);
    buf_zero4<<<(n4 + T - 1) / T, T, 0, stream>>>((float4*)AGG, n4);
    gcn_scatter<<<eblk, T, 0, stream>>>(H, AGG, src, dst, E, GCN_F_IN);
    gcn_finalize<<<(n4 + T - 1) / T, T, 0, stream>>>((const float4*)AGG, (const float4*)H,
                                                     dinv, (const float4*)b3, (float4*)OUT,
                                                     N, GCN_F_IN, 1);
  }
}